// Net_62285615727179
// MI455X (gfx1250) — compile-verified
//
#include <hip/hip_runtime.h>
#include <hip/hip_bf16.h>
#include <math.h>

#define N_NODES 100000
#define N_EDGES 1600000
#define TE_EDGES (N_EDGES + N_NODES)   // real edges + self loops

typedef __attribute__((ext_vector_type(2))) float v2f;
typedef __attribute__((ext_vector_type(8))) float v8f;

// ---------------------------------------------------------------- utilities

__global__ void k_fill(float* __restrict__ p, float v, long n) {
  long i = (long)blockIdx.x * blockDim.x + threadIdx.x;
  if (i < n) p[i] = v;
}

// dst[r, c] (Cd cols) = src[r, c] (Cs cols, Ks rows) zero-padded
__global__ void k_pad(const float* __restrict__ src, float* __restrict__ dst,
                      long Ks, int Cs, int Cd, long n) {
  long i = (long)blockIdx.x * blockDim.x + threadIdx.x;
  if (i >= n) return;
  long r = i / Cd;
  int  c = (int)(i % Cd);
  dst[i] = (r < Ks && c < Cs) ? src[r * Cs + c] : 0.0f;
}

__global__ void k_degree(const int* __restrict__ dst, float* __restrict__ deg) {
  int i = blockIdx.x * blockDim.x + threadIdx.x;
  if (i < N_EDGES) atomicAdd(&deg[dst[i]], 1.0f);
}

__global__ void k_dinv(const float* __restrict__ deg, float* __restrict__ dinv) {
  int i = blockIdx.x * blockDim.x + threadIdx.x;
  if (i < N_NODES) dinv[i] = rsqrtf(deg[i] + 1.0f);  // +1 for the self loop
}

__device__ __forceinline__ void edge_sd(int e, const int* __restrict__ src,
                                        const int* __restrict__ dst, int& s, int& d) {
  if (e < N_EDGES) { s = src[e]; d = dst[e]; }
  else             { s = e - N_EDGES; d = s; }        // self loop
}

// monotone float atomic max: signed-int max for non-negative, unsigned min for negative
__device__ __forceinline__ void atomicMaxF(float* addr, float val) {
  if (__float_as_int(val) >= 0)
    atomicMax((int*)addr, __float_as_int(val));
  else
    atomicMin((unsigned int*)addr, __float_as_uint(val));
}

__device__ __forceinline__ float leaky02(float e) { return e > 0.0f ? e : 0.2f * e; }

// ------------------------------------------------------- fp32 WMMA GEMM
// D[M x C] = A[M x K] * B[K x C], row-major, M%16==0, K%4==0, C%16==0 (by padding).
// 128-thread block (4 waves): block owns rows [rowBase, rowBase+64) x cols [tileN, tileN+16).
// A-block (64 x K, contiguous) and B-slice (K x 16) staged in LDS with float4 loads,
// then each wave runs an unrolled chain of V_WMMA_F32_16X16X4_F32 from ds_load fragments.
//
// ISA 7.12.2 layouts (wave32):
//   A 16x4 : lanes 0-15 row M=lane  hold {K=k, K=k+1}; lanes 16-31 hold {K=k+2, K=k+3}
//   B 4x16 : lanes 0-15 col N=lane  hold {K=k, K=k+1}; lanes 16-31 hold {K=k+2, K=k+3}
//   C/D    : VGPR v holds row tileM + v + 8*(lane>=16), col = tileN + (lane&15)
template <int K, int C>
__global__ void __launch_bounds__(128) k_gemm_wmma(
    const float* __restrict__ A, const float* __restrict__ B,
    float* __restrict__ D, int M) {
  __shared__ float ldsA[64 * K];
  __shared__ float ldsB[K * 16];

  const int tid  = threadIdx.x;
  const int lane = tid & 31;
  const int wave = tid >> 5;
  const int half = lane >> 4;      // 0: lanes 0-15, 1: lanes 16-31
  const int l16  = lane & 15;
  const int rowBase = blockIdx.x * 64;
  const int tileN   = blockIdx.y * 16;

  // stage A: contiguous 64*K floats starting at A + rowBase*K (guard only the grid tail)
  const long aBase  = (long)rowBase * K;
  const long aLimit = (long)M * K;
  for (int i = tid; i < (64 * K) / 4; i += 128) {
    long g = aBase + 4L * i;
    float4 v = make_float4(0.0f, 0.0f, 0.0f, 0.0f);
    if (g + 3 < aLimit) v = *(const float4*)(A + g);
    *(float4*)(&ldsA[4 * i]) = v;
  }
  // stage B slice: rows k=0..K-1, cols [tileN, tileN+16) — always in range
  for (int i = tid; i < (K * 16) / 4; i += 128) {
    int k = (4 * i) >> 4;
    int j = (4 * i) & 15;
    *(float4*)(&ldsB[4 * i]) = *(const float4*)(B + (long)k * C + tileN + j);
  }
  __syncthreads();

  v8f acc = {0.f, 0.f, 0.f, 0.f, 0.f, 0.f, 0.f, 0.f};
  const float* aRow = &ldsA[(wave * 16 + l16) * K];
#pragma unroll
  for (int k = 0; k < K; k += 4) {
    const int k0 = k + 2 * half;
    v2f av = *(const v2f*)(aRow + k0);
    v2f bv = {ldsB[k0 * 16 + l16], ldsB[(k0 + 1) * 16 + l16]};
    acc = __builtin_amdgcn_wmma_f32_16x16x4_f32(
        /*neg_a=*/false, av, /*neg_b=*/false, bv,
        /*c_mod=*/(short)0, acc, /*reuse_a=*/false, /*reuse_b=*/false);
  }

  const int tileM = rowBase + wave * 16;
  if (tileM < M) {                       // wave-uniform; M%16==0 so rows are all valid
    const int col = tileN + l16;
#pragma unroll
    for (int v = 0; v < 8; ++v)
      D[(long)(tileM + v + 8 * half) * C + col] = acc[v];
  }
}

// ---------------------------------------------------------------- GCN parts

template <int VEC>
__global__ void k_gcn_scatter(const float* __restrict__ H,
                              const int* __restrict__ src, const int* __restrict__ dst,
                              const float* __restrict__ dinv, float* __restrict__ out,
                              int C) {
  const int cpv = C / VEC;
  long tid = (long)blockIdx.x * blockDim.x + threadIdx.x;
  long total = (long)TE_EDGES * cpv;
  if (tid >= total) return;
  int e = (int)(tid / cpv);
  int c = (int)(tid % cpv) * VEC;
  int s, d; edge_sd(e, src, dst, s, d);
  float norm = dinv[s] * dinv[d];
  const float* hp = H + (long)s * C + c;
  float* op = out + (long)d * C + c;
#pragma unroll
  for (int j = 0; j < VEC; ++j) atomicAdd(op + j, norm * hp[j]);
}

__global__ void k_bias_relu(const float* __restrict__ agg, const float* __restrict__ bias,
                            float* __restrict__ out, int C, long n) {
  long i = (long)blockIdx.x * blockDim.x + threadIdx.x;
  if (i >= n) return;
  int c = (int)(i % C);
  float v = agg[i] + bias[c];
  out[i] = v > 0.0f ? v : 0.0f;
}

// ---------------------------------------------------------------- GAT parts

__global__ void k_alpha(const float* __restrict__ H, const float* __restrict__ a_s,
                        const float* __restrict__ a_d, float* __restrict__ asrc,
                        float* __restrict__ adst, int C, int ldH) {
  int i = blockIdx.x * blockDim.x + threadIdx.x;
  if (i >= N_NODES) return;
  const float* h = H + (long)i * ldH;
  float ss = 0.0f, sd = 0.0f;
  for (int c = 0; c < C; ++c) { float hv = h[c]; ss += hv * a_s[c]; sd += hv * a_d[c]; }
  asrc[i] = ss; adst[i] = sd;
}

__global__ void k_gat_max(const int* __restrict__ src, const int* __restrict__ dst,
                          const float* __restrict__ asrc, const float* __restrict__ adst,
                          float* __restrict__ mmax) {
  int e = blockIdx.x * blockDim.x + threadIdx.x;
  if (e >= TE_EDGES) return;
  int s, d; edge_sd(e, src, dst, s, d);
  atomicMaxF(&mmax[d], leaky02(asrc[s] + adst[d]));
}

__global__ void k_gat_expsum(const int* __restrict__ src, const int* __restrict__ dst,
                             const float* __restrict__ asrc, const float* __restrict__ adst,
                             const float* __restrict__ mmax, float* __restrict__ denom) {
  int e = blockIdx.x * blockDim.x + threadIdx.x;
  if (e >= TE_EDGES) return;
  int s, d; edge_sd(e, src, dst, s, d);
  float ev = leaky02(asrc[s] + adst[d]);
  atomicAdd(&denom[d], expf(ev - mmax[d]));
}

template <int VEC>
__global__ void k_gat_scatter(const float* __restrict__ H,
                              const int* __restrict__ src, const int* __restrict__ dst,
                              const float* __restrict__ asrc, const float* __restrict__ adst,
                              const float* __restrict__ mmax, const float* __restrict__ denom,
                              float* __restrict__ out, int C, int ldH) {
  const int cpv = C / VEC;
  long tid = (long)blockIdx.x * blockDim.x + threadIdx.x;
  long total = (long)TE_EDGES * cpv;
  if (tid >= total) return;
  int e = (int)(tid / cpv);
  int c = (int)(tid % cpv) * VEC;
  int s, d; edge_sd(e, src, dst, s, d);
  float ev = leaky02(asrc[s] + adst[d]);
  float w = expf(ev - mmax[d]) / (denom[d] + 1e-16f);
  const float* hp = H + (long)s * ldH + c;
  float* op = out + (long)d * C + c;
#pragma unroll
  for (int j = 0; j < VEC; ++j) atomicAdd(op + j, w * hp[j]);
}

// log_softmax over C=2, + bias, straight to output
__global__ void k_logsoftmax2(const float* __restrict__ agg, const float* __restrict__ bias,
                              float* __restrict__ out) {
  int i = blockIdx.x * blockDim.x + threadIdx.x;
  if (i >= N_NODES) return;
  float z0 = agg[2 * i + 0] + bias[0];
  float z1 = agg[2 * i + 1] + bias[1];
  float m = fmaxf(z0, z1);
  float lse = m + logf(expf(z0 - m) + expf(z1 - m));
  out[2 * i + 0] = z0 - lse;
  out[2 * i + 1] = z1 - lse;
}

// ---------------------------------------------------------------- launcher

static inline unsigned cdiv(long a, long b) { return (unsigned)((a + b - 1) / b); }

extern "C" void kernel_launch(void* const* d_in, const int* in_sizes, int n_in,
                              void* d_out, int out_size, void* d_ws, size_t ws_size,
                              hipStream_t stream) {
  (void)in_sizes; (void)n_in; (void)out_size; (void)ws_size;

  const float* x    = (const float*)d_in[0];
  const int*   ei   = (const int*)d_in[1];        // [2, E] flat
  const int*   src  = ei;
  const int*   dst  = ei + N_EDGES;
  const float* W1   = (const float*)d_in[2];
  const float* b1   = (const float*)d_in[3];
  const float* W2   = (const float*)d_in[4];
  const float* as2  = (const float*)d_in[5];
  const float* ad2  = (const float*)d_in[6];
  const float* b2   = (const float*)d_in[7];
  const float* W3   = (const float*)d_in[8];
  const float* b3   = (const float*)d_in[9];
  const float* W4   = (const float*)d_in[10];
  const float* as4  = (const float*)d_in[11];
  const float* ad4  = (const float*)d_in[12];
  const float* b4   = (const float*)d_in[13];
  float* out = (float*)d_out;

  // workspace layout (floats): 3 big N*128 buffers + scalars + padded operands (~160 MB)
  float* ws   = (float*)d_ws;
  const long NB = (long)N_NODES * 128;
  float* bufX  = ws;            // current layer input features  [N, Cprev]
  float* bufH  = ws + NB;       // linear-transformed features   [N, C] (ld = padded C)
  float* bufO  = ws + 2 * NB;   // aggregation accumulator       [N, C]
  float* deg   = ws + 3 * NB;
  float* dinv  = deg   + N_NODES;
  float* asrc  = dinv  + N_NODES;
  float* adst  = asrc  + N_NODES;
  float* mmax  = adst  + N_NODES;
  float* denom = mmax  + N_NODES;
  float* padX  = denom + N_NODES;        // N x 8 padded input features
  float* w1p   = padX  + (long)N_NODES * 8;  // 8 x 32
  float* w4p   = w1p   + 8 * 32;             // 128 x 16

  const int B = 256;
  const unsigned gemmGX = cdiv(N_NODES, 64);   // 64 rows per GEMM block

  // pad layer-1 input/weight (K 7->8) and layer-4 weight (C 2->16)
  k_pad<<<cdiv((long)N_NODES * 8, B), B, 0, stream>>>(x,  padX, N_NODES, 7, 8,  (long)N_NODES * 8);
  k_pad<<<cdiv(8 * 32, B),           B, 0, stream>>>(W1, w1p,  7,       32, 32, 8 * 32);
  k_pad<<<cdiv(128 * 16, B),         B, 0, stream>>>(W4, w4p,  128,     2,  16, 128 * 16);

  // degrees + symmetric norm (same edge set every layer)
  k_fill<<<cdiv(N_NODES, B), B, 0, stream>>>(deg, 0.0f, N_NODES);
  k_degree<<<cdiv(N_EDGES, B), B, 0, stream>>>(dst, deg);
  k_dinv<<<cdiv(N_NODES, B), B, 0, stream>>>(deg, dinv);

  // ---------------- Layer 1: GCN 7 -> 32 ----------------
  {
    const int C = 32; const long NC = (long)N_NODES * C;
    k_gemm_wmma<8, 32><<<dim3(gemmGX, C / 16), 128, 0, stream>>>(padX, w1p, bufH, N_NODES);
    k_fill<<<cdiv(NC, B), B, 0, stream>>>(bufO, 0.0f, NC);
    k_gcn_scatter<4><<<cdiv((long)TE_EDGES * (C / 4), B), B, 0, stream>>>(bufH, src, dst, dinv, bufO, C);
    k_bias_relu<<<cdiv(NC, B), B, 0, stream>>>(bufO, b1, bufX, C, NC);
  }
  // ---------------- Layer 2: GAT 32 -> 64 ----------------
  {
    const int C = 64; const long NC = (long)N_NODES * C;
    k_gemm_wmma<32, 64><<<dim3(gemmGX, C / 16), 128, 0, stream>>>(bufX, W2, bufH, N_NODES);
    k_alpha<<<cdiv(N_NODES, B), B, 0, stream>>>(bufH, as2, ad2, asrc, adst, C, C);
    k_fill<<<cdiv(N_NODES, B), B, 0, stream>>>(mmax, -INFINITY, N_NODES);
    k_gat_max<<<cdiv(TE_EDGES, B), B, 0, stream>>>(src, dst, asrc, adst, mmax);
    k_fill<<<cdiv(N_NODES, B), B, 0, stream>>>(denom, 0.0f, N_NODES);
    k_gat_expsum<<<cdiv(TE_EDGES, B), B, 0, stream>>>(src, dst, asrc, adst, mmax, denom);
    k_fill<<<cdiv(NC, B), B, 0, stream>>>(bufO, 0.0f, NC);
    k_gat_scatter<4><<<cdiv((long)TE_EDGES * (C / 4), B), B, 0, stream>>>(
        bufH, src, dst, asrc, adst, mmax, denom, bufO, C, C);
    k_bias_relu<<<cdiv(NC, B), B, 0, stream>>>(bufO, b2, bufX, C, NC);
  }
  // ---------------- Layer 3: GCN 64 -> 128 ----------------
  {
    const int C = 128; const long NC = (long)N_NODES * C;
    k_gemm_wmma<64, 128><<<dim3(gemmGX, C / 16), 128, 0, stream>>>(bufX, W3, bufH, N_NODES);
    k_fill<<<cdiv(NC, B), B, 0, stream>>>(bufO, 0.0f, NC);
    k_gcn_scatter<4><<<cdiv((long)TE_EDGES * (C / 4), B), B, 0, stream>>>(bufH, src, dst, dinv, bufO, C);
    k_bias_relu<<<cdiv(NC, B), B, 0, stream>>>(bufO, b3, bufX, C, NC);
  }
  // ---------------- Layer 4: GAT 128 -> 2 (padded C=16) + log_softmax ----------------
  {
    const int C = 2, LD = 16; const long NC = (long)N_NODES * C;
    k_gemm_wmma<128, 16><<<dim3(gemmGX, 1), 128, 0, stream>>>(bufX, w4p, bufH, N_NODES);
    k_alpha<<<cdiv(N_NODES, B), B, 0, stream>>>(bufH, as4, ad4, asrc, adst, C, LD);
    k_fill<<<cdiv(N_NODES, B), B, 0, stream>>>(mmax, -INFINITY, N_NODES);
    k_gat_max<<<cdiv(TE_EDGES, B), B, 0, stream>>>(src, dst, asrc, adst, mmax);
    k_fill<<<cdiv(N_NODES, B), B, 0, stream>>>(denom, 0.0f, N_NODES);
    k_gat_expsum<<<cdiv(TE_EDGES, B), B, 0, stream>>>(src, dst, asrc, adst, mmax, denom);
    k_fill<<<cdiv(NC, B), B, 0, stream>>>(bufO, 0.0f, NC);
    k_gat_scatter<2><<<cdiv((long)TE_EDGES * (C / 2), B), B, 0, stream>>>(
        bufH, src, dst, asrc, adst, mmax, denom, bufO, C, LD);
    k_logsoftmax2<<<cdiv(N_NODES, B), B, 0, stream>>>(bufO, b4, out);
  }
}